// GNNDecoder_50036368998578
// MI455X (gfx1250) — compile-verified
//
#include <hip/hip_runtime.h>
#include <hip/hip_bf16.h>
#include <math.h>

#define N0_NODES 100000
#define N1_NODES 50000
#define N2_NODES 25000
#define E0_EDGES 1600000
#define E1_EDGES 800000
#define D_LAT 64
#define D_H 128
#define BN_EPS 1e-5f

typedef __attribute__((ext_vector_type(2))) float v2f;
typedef __attribute__((ext_vector_type(8))) float v8f;

static inline int cdiv(long long a, long long b) { return (int)((a + b - 1) / b); }

// ---------------------------------------------------------------- utilities
__global__ void fill_f32(float* __restrict__ p, float v, int n) {
    int i = blockIdx.x * blockDim.x + threadIdx.x;
    if (i < n) p[i] = v;
}

// xout[perm[r]] = xin[r]   (rows of f4_per_row float4's; xout pre-zeroed)
__global__ void scatter_rows_f4(const float* __restrict__ xin, const int* __restrict__ perm,
                                float* __restrict__ xout, int nrows, int f4_per_row) {
    int i = blockIdx.x * blockDim.x + threadIdx.x;
    int total = nrows * f4_per_row;
    if (i >= total) return;
    int r = i / f4_per_row, c = i - r * f4_per_row;
    int dr = perm[r];
    ((float4*)xout)[(size_t)dr * f4_per_row + c] =
        ((const float4*)xin)[(size_t)r * f4_per_row + c];
}

// deg[dst[e]] += w[e]  (deg pre-initialized to 1.0 == self-loop weight)
__global__ void edge_degree(const int* __restrict__ dst, const float* __restrict__ w,
                            float* __restrict__ deg, int E) {
    int e = blockIdx.x * blockDim.x + threadIdx.x;
    if (e < E) atomicAdd(&deg[dst[e]], w[e]);
}

__global__ void deg_to_dinv(float* __restrict__ deg, int n) {
    int i = blockIdx.x * blockDim.x + threadIdx.x;
    if (i < n) { float d = deg[i]; deg[i] = d > 0.f ? rsqrtf(d) : 0.f; }
}

// ------------------------------------------------ WMMA fp32 GEMM: OUT = X*W
// X: [M x K] row-major, W: [K x 128] row-major, OUT: [M x 128] row-major.
// Block = 256 threads (8 waves): one 16-row stripe of OUT; wave w -> N-tile w.
// Uses V_WMMA_F32_16X16X4_F32 (full f32 precision, matches reference math).
template <int K>
__global__ void gemm_wmma_f32(const float* __restrict__ X, const float* __restrict__ W,
                              float* __restrict__ OUT) {
    __shared__ float As[16 * K];
    const int m0 = blockIdx.x * 16;
    // cooperative load of the 16xK A tile into LDS (A reused by all 8 waves)
    for (int idx = threadIdx.x; idx < 16 * K / 4; idx += 256)
        ((float4*)As)[idx] = ((const float4*)(X + (size_t)m0 * K))[idx];
    __syncthreads();

    const int wave = threadIdx.x >> 5;
    const int lane = threadIdx.x & 31;
    const int half = lane >> 4;   // 0: K pair {k,k+1} / M rows 0-7 ; 1: {k+2,k+3} / rows 8-15
    const int l16  = lane & 15;
    const int n0   = wave * 16;

    v8f c = {};
    for (int k = 0; k < K; k += 4) {
        v2f a, b;
        // A 16x4 layout: lanes 0-15 hold (K=k,k+1), lanes 16-31 hold (K=k+2,k+3), M = lane&15
        a.x = As[l16 * K + k + 2 * half];
        a.y = As[l16 * K + k + 2 * half + 1];
        // B 4x16 layout (row striped across lanes): VGPR0 rows k / k+2, VGPR1 rows k+1 / k+3
        b.x = W[(size_t)(k + 2 * half) * D_H + n0 + l16];
        b.y = W[(size_t)(k + 2 * half + 1) * D_H + n0 + l16];
        c = __builtin_amdgcn_wmma_f32_16x16x4_f32(
                /*neg_a=*/false, a, /*neg_b=*/false, b,
                /*c_mod=*/(short)0, c, /*reuse_a=*/false, /*reuse_b=*/false);
    }
    // C/D layout: VGPR j -> M = j (lanes 0-15) / j+8 (lanes 16-31), N = lane&15
#pragma unroll
    for (int j = 0; j < 8; ++j)
        OUT[(size_t)(m0 + j + 8 * half) * D_H + n0 + l16] = c[j];
}

// -------------------------------------------- edge aggregation (wave / edge)
__global__ void edge_aggregate(const int* __restrict__ src, const int* __restrict__ dst,
                               const float* __restrict__ w, const float* __restrict__ dinv,
                               const float* __restrict__ h, float* __restrict__ out, int E) {
    int e = blockIdx.x * (blockDim.x >> 5) + (threadIdx.x >> 5);
    if (e >= E) return;
    int lane = threadIdx.x & 31;
    int s = src[e], d = dst[e];
    float norm = dinv[s] * w[e] * dinv[d];
    const float* hrow = h + (size_t)s * D_H;
    float* orow = out + (size_t)d * D_H;
#pragma unroll
    for (int c = lane; c < D_H; c += 32)
        atomicAdd(&orow[c], norm * hrow[c]);
}

// self-loop contribution: out[i] += dinv[i]^2 * h[i]   (weight 1 self-loop)
__global__ void self_aggregate(const float* __restrict__ dinv, const float* __restrict__ h,
                               float* __restrict__ out, int n) {
    int i = blockIdx.x * blockDim.x + threadIdx.x;
    if (i >= n * D_H) return;
    int r = i >> 7;
    float di = dinv[r];
    out[i] += di * di * h[i];
}

// ------------------------------------------------------------ BatchNorm
// stats[0..127] = sum per channel, stats[128..255] = sum of squares
__global__ void bn_stats(const float* __restrict__ x, float* __restrict__ stats, int n) {
    __shared__ float ssum[256], ssq[256];
    int c = threadIdx.x & 127;
    float s = 0.f, q = 0.f;
    for (int r = blockIdx.x * 2 + (threadIdx.x >> 7); r < n; r += gridDim.x * 2) {
        float v = x[(size_t)r * D_H + c];
        s += v; q += v * v;
    }
    ssum[threadIdx.x] = s; ssq[threadIdx.x] = q;
    __syncthreads();
    if (threadIdx.x < 128) {
        atomicAdd(&stats[c],       ssum[threadIdx.x] + ssum[threadIdx.x + 128]);
        atomicAdd(&stats[128 + c], ssq[threadIdx.x] + ssq[threadIdx.x + 128]);
    }
}

__global__ void bn_apply_relu(float* __restrict__ x, const float* __restrict__ stats,
                              const float* __restrict__ gamma, const float* __restrict__ beta,
                              int n) {
    int i = blockIdx.x * blockDim.x + threadIdx.x;
    if (i >= n * D_H) return;
    int c = i & 127;
    float inv_n = 1.f / (float)n;
    float mean = stats[c] * inv_n;
    float var  = stats[128 + c] * inv_n - mean * mean;
    float v = (x[i] - mean) * rsqrtf(var + BN_EPS) * gamma[c] + beta[c];
    x[i] = v > 0.f ? v : 0.f;
}

// -------------------------------------------------- gaussian head (wave/node)
__global__ void head_kernel(const float* __restrict__ y, const float* __restrict__ Wf,
                            const float* __restrict__ bf, float* __restrict__ out, int n) {
    int node = blockIdx.x * (blockDim.x >> 5) + (threadIdx.x >> 5);
    if (node >= n) return;
    int lane = threadIdx.x & 31;
    const float* row = y + (size_t)node * D_H;
    float a0 = 0.f, a1 = 0.f;
#pragma unroll
    for (int j = 0; j < 4; ++j) {
        int k = lane * 4 + j;              // wave32: 32 lanes x 4 = 128 features
        float v = row[k];
        a0 += v * Wf[k * 2 + 0];
        a1 += v * Wf[k * 2 + 1];
    }
#pragma unroll
    for (int off = 16; off > 0; off >>= 1) {
        a0 += __shfl_xor(a0, off, 32);
        a1 += __shfl_xor(a1, off, 32);
    }
    if (lane == 0) {
        float mu = a0 + bf[0];
        float p1 = a1 + bf[1];
        float sp = p1 > 20.f ? p1 : log1pf(expf(p1));   // stable softplus
        out[(size_t)node * 2 + 0] = mu;
        out[(size_t)node * 2 + 1] = sp + 1e-6f;
    }
}

// ---------------------------------------------------------------------------
extern "C" void kernel_launch(void* const* d_in, const int* in_sizes, int n_in,
                              void* d_out, int out_size, void* d_ws, size_t ws_size,
                              hipStream_t stream) {
    (void)in_sizes; (void)n_in; (void)ws_size;
    const float* x_latent = (const float*)d_in[0];
    const int*   perm1    = (const int*)d_in[1];
    const int*   perm0    = (const int*)d_in[2];
    const int*   ei1      = (const int*)d_in[3];   // [2, E1]: src then dst
    const float* ew1      = (const float*)d_in[4];
    const int*   ei0      = (const int*)d_in[5];   // [2, E0]
    const float* ew0      = (const float*)d_in[6];
    // d_in[7] = batch0 (zeros), d_in[9]/d_in[13] = b0/b1 (cancelled by BN mean-subtraction)
    const float* W0 = (const float*)d_in[8];
    const float* g0 = (const float*)d_in[10];
    const float* be0 = (const float*)d_in[11];
    const float* W1 = (const float*)d_in[12];
    const float* g1 = (const float*)d_in[14];
    const float* be1 = (const float*)d_in[15];
    const float* Wf = (const float*)d_in[16];
    const float* bf = (const float*)d_in[17];

    float* buf0  = (float*)d_ws;                            // unpooled X
    float* buf1  = buf0 + (size_t)N0_NODES * D_H;           // H = X*W
    float* buf2  = buf1 + (size_t)N0_NODES * D_H;           // aggregated / BN-ReLU
    float* deg   = buf2 + (size_t)N0_NODES * D_H;           // degree -> d^-1/2
    float* stats = deg + N0_NODES;                          // 256 BN partials
    float* out   = (float*)d_out;

    const int T = 256;
    // =================== decoder layer 0: N2 -> N1, GCN(64->128), BN, ReLU
    fill_f32<<<cdiv((long long)N1_NODES * D_LAT, T), T, 0, stream>>>(buf0, 0.f, N1_NODES * D_LAT);
    scatter_rows_f4<<<cdiv((long long)N2_NODES * 16, T), T, 0, stream>>>(x_latent, perm1, buf0, N2_NODES, D_LAT / 4);
    fill_f32<<<cdiv(N1_NODES, T), T, 0, stream>>>(deg, 1.f, N1_NODES);
    edge_degree<<<cdiv(E1_EDGES, T), T, 0, stream>>>(ei1 + E1_EDGES, ew1, deg, E1_EDGES);
    deg_to_dinv<<<cdiv(N1_NODES, T), T, 0, stream>>>(deg, N1_NODES);
    gemm_wmma_f32<D_LAT><<<N1_NODES / 16, T, 0, stream>>>(buf0, W0, buf1);
    fill_f32<<<cdiv((long long)N1_NODES * D_H, T), T, 0, stream>>>(buf2, 0.f, N1_NODES * D_H);
    edge_aggregate<<<cdiv(E1_EDGES, 8), T, 0, stream>>>(ei1, ei1 + E1_EDGES, ew1, deg, buf1, buf2, E1_EDGES);
    self_aggregate<<<cdiv((long long)N1_NODES * D_H, T), T, 0, stream>>>(deg, buf1, buf2, N1_NODES);
    fill_f32<<<1, T, 0, stream>>>(stats, 0.f, 256);
    bn_stats<<<1024, T, 0, stream>>>(buf2, stats, N1_NODES);
    bn_apply_relu<<<cdiv((long long)N1_NODES * D_H, T), T, 0, stream>>>(buf2, stats, g0, be0, N1_NODES);

    // =================== decoder layer 1: N1 -> N0, GCN(128->128), BN, ReLU
    fill_f32<<<cdiv((long long)N0_NODES * D_H, T), T, 0, stream>>>(buf0, 0.f, N0_NODES * D_H);
    scatter_rows_f4<<<cdiv((long long)N1_NODES * 32, T), T, 0, stream>>>(buf2, perm0, buf0, N1_NODES, D_H / 4);
    fill_f32<<<cdiv(N0_NODES, T), T, 0, stream>>>(deg, 1.f, N0_NODES);
    edge_degree<<<cdiv(E0_EDGES, T), T, 0, stream>>>(ei0 + E0_EDGES, ew0, deg, E0_EDGES);
    deg_to_dinv<<<cdiv(N0_NODES, T), T, 0, stream>>>(deg, N0_NODES);
    gemm_wmma_f32<D_H><<<N0_NODES / 16, T, 0, stream>>>(buf0, W1, buf1);
    fill_f32<<<cdiv((long long)N0_NODES * D_H, T), T, 0, stream>>>(buf2, 0.f, N0_NODES * D_H);
    edge_aggregate<<<cdiv(E0_EDGES, 8), T, 0, stream>>>(ei0, ei0 + E0_EDGES, ew0, deg, buf1, buf2, E0_EDGES);
    self_aggregate<<<cdiv((long long)N0_NODES * D_H, T), T, 0, stream>>>(deg, buf1, buf2, N0_NODES);
    fill_f32<<<1, T, 0, stream>>>(stats, 0.f, 256);
    bn_stats<<<1024, T, 0, stream>>>(buf2, stats, N0_NODES);
    bn_apply_relu<<<cdiv((long long)N0_NODES * D_H, T), T, 0, stream>>>(buf2, stats, g1, be1, N0_NODES);

    // =================== gaussian head + batch0 (zeros) tail
    head_kernel<<<cdiv(N0_NODES, 8), T, 0, stream>>>(buf2, Wf, bf, out, N0_NODES);
    if (out_size > 2 * N0_NODES)
        fill_f32<<<cdiv(out_size - 2 * N0_NODES, T), T, 0, stream>>>(out + 2 * N0_NODES, 0.f, out_size - 2 * N0_NODES);
}